// GaussianConv2DAtScale_50319836839995
// MI455X (gfx1250) — compile-verified
//
#include <hip/hip_runtime.h>

typedef __attribute__((ext_vector_type(2))) float v2f;
typedef __attribute__((ext_vector_type(8))) float v8f;

#define HW    512
#define CHN   32
#define TS    16
#define RAD   3
#define INR   22            // TS + 2*RAD input rows
#define INC   22            // TS + 2*RAD input cols
#define RSTR  25            // LDS row stride (floats, odd -> bank friendly)
#define CSTR  553           // LDS per-channel stride = 22*25 + 3 (odd)
#define WOFF  (CSTR*CHN)    // 17696 floats: weights region
#define WCNT  (49*CHN)      // 1568 floats
#define OOFF  (WOFF+WCNT)   // 19264 floats: output staging region
#define OPIX  33            // per-pixel channel stride in out staging (odd)
#define SMEMF (OOFF + TS*TS*OPIX)   // 27712 floats = 110848 bytes

// generic->LDS byte offset: flat LDS aperture uses addr[31:0] as LDS address
__device__ __forceinline__ unsigned lds_byte_off(const void* p) {
    return (unsigned)(unsigned long long)p;
}

__global__ __launch_bounds__(256) void gauss_dwconv_wmma(
    const float* __restrict__ xin, const float* __restrict__ wgt,
    const float* __restrict__ bias, float* __restrict__ out)
{
    extern __shared__ float smem[];

    const int tid  = threadIdx.x;
    const int wv   = tid >> 5;      // wave id 0..7
    const int lane = tid & 31;
    const int hl   = lane >> 4;     // half-wave select
    const int l16  = lane & 15;

    const int tx = blockIdx.x, ty = blockIdx.y, nb = blockIdx.z;
    const int x0 = tx * TS, y0 = ty * TS;

    // ---- Phase 0: zero input slab (handles SAME padding + pad columns) ----
    {
        float4* z = (float4*)smem;
        float4 zero4 = make_float4(0.f, 0.f, 0.f, 0.f);
        for (int i = tid; i < WOFF / 4; i += 256) z[i] = zero4;
    }
    // stage weights: layout already [ky][kx][c]
    for (int i = tid; i < WCNT; i += 256) smem[WOFF + i] = wgt[i];
    __syncthreads();

    // ---- Phase 1: async global -> LDS halo load. lane == channel ----
    for (int p = wv; p < INR * INC; p += 8) {
        const int r   = p / INC;
        const int cgl = p - r * INC;
        const int gy  = y0 - RAD + r;
        const int gx  = x0 - RAD + cgl;
        if ((unsigned)gy < HW && (unsigned)gx < HW) {   // wave-uniform predicate
            unsigned laddr = lds_byte_off(&smem[lane * CSTR + r * RSTR + cgl]);
            const float* gp = xin + ((((size_t)nb * HW + gy) * HW + gx) * CHN + lane);
            asm volatile("global_load_async_to_lds_b32 %0, %1, off"
                         :: "v"(laddr), "v"(gp) : "memory");
        }
    }
    asm volatile("s_wait_asynccnt 0" ::: "memory");
    __syncthreads();

    // ---- Phase 2: per-channel banded-matmul conv on the matrix pipe ----
    for (int cc = 0; cc < 4; ++cc) {
        const int c  = wv * 4 + cc;
        const int cb = c * CSTR;

        v8f acc = (v8f){0.f, 0.f, 0.f, 0.f, 0.f, 0.f, 0.f, 0.f};

        for (int ky = 0; ky < 7; ++ky) {
            // Build Toeplitz B chunks for this ky.
            // B 4x16 layout: VGPR0 -> K = 2*hl, VGPR1 -> K = 2*hl+1; N = l16.
            v2f Bch[6];
            #pragma unroll
            for (int k = 0; k < 6; ++k) {
                #pragma unroll
                for (int e = 0; e < 2; ++e) {
                    const int K   = 2 * hl + e;
                    const int l   = 4 * k + K;       // lds column index
                    const int tap = l - l16;         // kx
                    float v = 0.0f;
                    if (tap >= 0 && tap < 7)
                        v = smem[WOFF + (ky * 7 + tap) * CHN + c];
                    Bch[k][e] = v;
                }
            }
            // A 16x4 layout: M = l16, K = 2*hl + e (contiguous lds columns)
            const int arow = cb + (l16 + ky) * RSTR;
            #pragma unroll
            for (int k = 0; k < 6; ++k) {
                v2f A;
                const int abase = arow + 4 * k + 2 * hl;
                A[0] = smem[abase + 0];
                A[1] = smem[abase + 1];
                acc = __builtin_amdgcn_wmma_f32_16x16x4_f32(
                          false, A, false, Bch[k],
                          (short)0, acc, false, false);
            }
        }

        // epilogue: relu(acc + b[c]) * sigma^2, staged to LDS
        const float bc = bias[c];
        #pragma unroll
        for (int j = 0; j < 8; ++j) {
            const float v   = fmaxf(acc[j] + bc, 0.0f) * 2.25f;
            const int   row = j + 8 * hl;         // C/D layout: M = j (+8 for hi half)
            const int   col = l16;                // N = lane
            smem[OOFF + (row * TS + col) * OPIX + c] = v;
        }
    }
    __syncthreads();

    // ---- Phase 3: coalesced b128 stores of the 16x16x32 tile ----
    for (int i = tid; i < TS * TS * 8; i += 256) {
        const int pix = i >> 3;
        const int cq  = (i & 7) * 4;
        const int row = pix >> 4, col = pix & 15;
        float4 v;
        v.x = smem[OOFF + pix * OPIX + cq + 0];
        v.y = smem[OOFF + pix * OPIX + cq + 1];
        v.z = smem[OOFF + pix * OPIX + cq + 2];
        v.w = smem[OOFF + pix * OPIX + cq + 3];
        const size_t o = (((size_t)nb * HW + (y0 + row)) * HW + (x0 + col)) * CHN + cq;
        *(float4*)(out + o) = v;
    }
}

extern "C" void kernel_launch(void* const* d_in, const int* in_sizes, int n_in,
                              void* d_out, int out_size, void* d_ws, size_t ws_size,
                              hipStream_t stream) {
    const float* x = (const float*)d_in[0];   // (8,512,512,32)
    const float* W = (const float*)d_in[1];   // (7,7,1,32)
    const float* b = (const float*)d_in[2];   // (32,)
    float* out = (float*)d_out;

    dim3 grid(HW / TS, HW / TS, 8);
    dim3 block(256);
    size_t shmem = (size_t)SMEMF * sizeof(float);
    gauss_dwconv_wmma<<<grid, block, shmem, stream>>>(x, W, b, out);
}